// GNN_13511967113638
// MI455X (gfx1250) — compile-verified
//
#include <hip/hip_runtime.h>

#define NN 100000
#define EE 1600000
#define DD 64
#define LL 3
#define BN_EPS 1e-5f

typedef __attribute__((ext_vector_type(2))) float v2f;
typedef __attribute__((ext_vector_type(4))) float v4f;
typedef __attribute__((ext_vector_type(8))) float v8f;

// ---------------------------------------------------------------- utilities
__global__ void zero_f4(v4f* __restrict__ p, int n4) {
    int i = blockIdx.x * blockDim.x + threadIdx.x;
    if (i < n4) p[i] = (v4f){0.f, 0.f, 0.f, 0.f};
}

__global__ void degree_count(const int* __restrict__ dst, float* __restrict__ deg, int e) {
    int i = blockIdx.x * blockDim.x + threadIdx.x;
    if (i < e) atomicAdd(deg + dst[i], 1.0f);
}

__global__ void degree_invert(float* __restrict__ deg, int n) {
    int i = blockIdx.x * blockDim.x + threadIdx.x;
    if (i < n) deg[i] = 1.0f / fmaxf(deg[i], 1.0f);
}

// ------------------------------------------------- mean-aggregation scatter
// 16 threads per edge; each handles 4 consecutive features (float4 gather is
// coalesced across the 16-thread group; scatter via L2-resident f32 atomics).
__global__ void sage_aggregate(const float* __restrict__ h,
                               const int* __restrict__ src,
                               const int* __restrict__ dst,
                               float* __restrict__ hn) {
    int t = blockIdx.x * blockDim.x + threadIdx.x;   // EE*16 threads total
    int e = t >> 4;
    if (e >= EE) return;
    int c = (t & 15) << 2;
    int s = src[e], d = dst[e];
    v4f v = *(const v4f*)(h + (size_t)s * DD + c);
    float* p = hn + (size_t)d * DD + c;
    atomicAdd(p + 0, v.x);
    atomicAdd(p + 1, v.y);
    atomicAdd(p + 2, v.z);
    atomicAdd(p + 3, v.w);
}

// --------------------------------------- fused SAGE GEMM + bias + BN + ReLU
// hout = relu( BN( h @ Ws + (hn * dinv) @ Wn + bias ) )
// One wave computes a 16x64 strip: 4 independent 16x16 f32 accumulators fed
// by V_WMMA_F32_16X16X4_F32. Weights are staged once per block into LDS,
// pre-swizzled into the exact per-lane B-fragment layout so the inner loop
// is one conflict-free ds_load_b64 per WMMA. Block = 8 waves = 128 rows.
__global__ __launch_bounds__(256) void sage_gemm_bn_relu(
    const float* __restrict__ h,
    const float* __restrict__ hn,
    const float* __restrict__ dinv,
    const float* __restrict__ Ws,     // 64x64 row-major [k][n]
    const float* __restrict__ Wn,     // 64x64 row-major [k][n]
    const float* __restrict__ bias,   // 64
    const float* __restrict__ gamma,  // 64
    const float* __restrict__ beta,   // 64
    const float* __restrict__ mean,   // 64
    const float* __restrict__ var,    // 64
    float* __restrict__ hout)
{
    // B-fragments: [kstep 0..15][tile 0..3][lane 0..31] as v2f.
    // Lane L of tile t at kstep ks holds { W[ka][c], W[ka+1][c] } with
    // ka = 4*ks + 2*(L>=16), c = 16*t + (L&15)  (ISA 7.12.2 B layout).
    __shared__ v2f fragS[16 * 128];   // 16 KB
    __shared__ v2f fragN[16 * 128];   // 16 KB

    for (int i = threadIdx.x; i < 16 * 128; i += 256) {
        const int ks  = i >> 7;
        const int rem = i & 127;
        const int t   = rem >> 5;
        const int ln  = rem & 31;
        const int ka  = ks * 4 + ((ln >> 4) << 1);
        const int c   = t * 16 + (ln & 15);
        fragS[i] = (v2f){ Ws[(ka + 0) * DD + c], Ws[(ka + 1) * DD + c] };
        fragN[i] = (v2f){ Wn[(ka + 0) * DD + c], Wn[(ka + 1) * DD + c] };
    }
    __syncthreads();

    const int lane = threadIdx.x & 31;
    const int wave = threadIdx.x >> 5;                 // 0..7
    const int tile = blockIdx.x * 8 + wave;            // 16-row tile index
    if (tile >= NN / 16) return;                       // whole-wave exit (after barrier)

    const int row0 = tile * 16;
    const int half = lane >> 4;                        // K-half select (A layout)
    const int l    = lane & 15;
    const int arow = row0 + l;                         // A-matrix row for this lane

    const float dw     = dinv[arow];
    const float* hrow  = h  + (size_t)arow * DD;
    const float* hnrow = hn + (size_t)arow * DD;

    v8f acc[4] = {};

    #pragma unroll
    for (int ks = 0; ks < 16; ++ks) {
        const int ka = ks * 4 + 2 * half;
        v2f a  = *(const v2f*)(hrow + ka);
        v2f an = *(const v2f*)(hnrow + ka);
        an.x *= dw; an.y *= dw;
        const v2f* fS = fragS + ks * 128 + lane;
        const v2f* fN = fragN + ks * 128 + lane;
        #pragma unroll
        for (int t = 0; t < 4; ++t) {
            acc[t] = __builtin_amdgcn_wmma_f32_16x16x4_f32(
                false, a,  false, fS[t * 32], (short)0, acc[t], false, false);
            acc[t] = __builtin_amdgcn_wmma_f32_16x16x4_f32(
                false, an, false, fN[t * 32], (short)0, acc[t], false, false);
        }
    }

    // epilogue: bias + BatchNorm(eval) + ReLU (per-column constants)
    const int rbase = row0 + 8 * half;                 // C/D layout: M = r + 8*half
    #pragma unroll
    for (int t = 0; t < 4; ++t) {
        const int col = t * 16 + l;
        const float bb = bias[col];
        const float mu = mean[col];
        const float sc = gamma[col] * rsqrtf(var[col] + BN_EPS);
        const float be = beta[col];
        #pragma unroll
        for (int r = 0; r < 8; ++r) {
            float v = acc[t][r] + bb;
            v = (v - mu) * sc + be;
            v = v > 0.f ? v : 0.f;
            hout[(size_t)(rbase + r) * DD + col] = v;
        }
    }
}

// ------------------------------------------------------- final classifier
__global__ void classifier_kernel(const float* __restrict__ h,
                                  const float* __restrict__ Wc,   // 64x2
                                  const float* __restrict__ bc,   // 2
                                  float* __restrict__ out) {
    int i = blockIdx.x * blockDim.x + threadIdx.x;
    if (i >= NN) return;
    const float* row = h + (size_t)i * DD;
    float a0 = bc[0], a1 = bc[1];
    #pragma unroll 8
    for (int k = 0; k < DD; ++k) {
        float x = row[k];
        a0 = fmaf(x, Wc[k * 2 + 0], a0);
        a1 = fmaf(x, Wc[k * 2 + 1], a1);
    }
    out[(size_t)i * 2 + 0] = a0;
    out[(size_t)i * 2 + 1] = a1;
}

// ----------------------------------------------------------------- launcher
extern "C" void kernel_launch(void* const* d_in, const int* in_sizes, int n_in,
                              void* d_out, int out_size, void* d_ws, size_t ws_size,
                              hipStream_t stream) {
    (void)in_sizes; (void)n_in; (void)out_size; (void)ws_size;

    const float* features = (const float*)d_in[0];
    const int*   src      = (const int*)  d_in[1];
    const int*   dst      = (const int*)  d_in[2];
    const float* W_self   = (const float*)d_in[3];
    const float* W_neigh  = (const float*)d_in[4];
    const float* b        = (const float*)d_in[5];
    const float* gamma    = (const float*)d_in[6];
    const float* beta     = (const float*)d_in[7];
    const float* mean     = (const float*)d_in[8];
    const float* var      = (const float*)d_in[9];
    const float* Wc       = (const float*)d_in[10];
    const float* bc       = (const float*)d_in[11];
    float* out = (float*)d_out;

    // workspace layout: hA | hB | hn | deg  = (3*64+1)*N floats (~77 MB)
    float* hA  = (float*)d_ws;
    float* hB  = hA + (size_t)NN * DD;
    float* hn  = hB + (size_t)NN * DD;
    float* deg = hn + (size_t)NN * DD;

    // degree -> inverse degree (recomputed each launch: no state carryover)
    zero_f4<<<(NN / 4 + 255) / 256, 256, 0, stream>>>((v4f*)deg, NN / 4);
    degree_count<<<(EE + 255) / 256, 256, 0, stream>>>(dst, deg, EE);
    degree_invert<<<(NN + 255) / 256, 256, 0, stream>>>(deg, NN);

    const int gemm_grid = (NN / 16 + 7) / 8;   // 782 blocks of 8 waves
    const float* hin = features;
    float* pong[2] = {hA, hB};
    for (int layer = 0; layer < LL; ++layer) {
        float* hout = pong[layer & 1];
        zero_f4<<<(NN * DD / 4 + 255) / 256, 256, 0, stream>>>((v4f*)hn, NN * DD / 4);
        sage_aggregate<<<(EE * 16) / 256, 256, 0, stream>>>(hin, src, dst, hn);
        sage_gemm_bn_relu<<<gemm_grid, 256, 0, stream>>>(
            hin, hn, deg,
            W_self + (size_t)layer * DD * DD,
            W_neigh + (size_t)layer * DD * DD,
            b + layer * DD, gamma + layer * DD, beta + layer * DD,
            mean + layer * DD, var + layer * DD, hout);
        hin = hout;
    }
    classifier_kernel<<<(NN + 255) / 256, 256, 0, stream>>>(hin, Wc, bc, out);
}